// NeuralExecutionModule_51591147159726
// MI455X (gfx1250) — compile-verified
//
#include <hip/hip_runtime.h>
#include <math.h>

// ---------------------------------------------------------------------------
// NeuralExecutionModule for MI455X (gfx1250, wave32, WMMA).
// Phase A: logits = z @ W_*          (fp32 WMMA 16x16x4, 16x64 tile / wave,
//                                     branch-free inner loop via clamped loads)
// Phase B: soft-VM decode+scan       (wave32: lane == register index)
// Phase C: Rf @ W_r2h^T + LayerNorm  (fp32 WMMA, fused LN in LDS)
// ---------------------------------------------------------------------------

typedef __attribute__((ext_vector_type(2))) float v2f;
typedef __attribute__((ext_vector_type(8))) float v8f;

#define B_SZ  4096
#define HIDD  512
#define NREG  32
#define NBIT  8
#define TLEN  32
#define OPB   2
#define AB    5
#define LB    5
#define GNUM  8

__device__ __forceinline__ float wred_sum(float x) {
  // full-wave (32 lane) butterfly sum; result broadcast to all lanes
  x += __shfl_xor(x, 16, 32);
  x += __shfl_xor(x, 8, 32);
  x += __shfl_xor(x, 4, 32);
  x += __shfl_xor(x, 2, 32);
  x += __shfl_xor(x, 1, 32);
  return x;
}

__device__ __forceinline__ float sigmoidf(float x) {
  return 1.0f / (1.0f + __expf(-x));
}

// ---------------------------------------------------------------------------
// Phase A: C[M,N] = A[M,512] * W[512,N], all row-major fp32.
// Wave computes a 16(M) x 64(N) tile with 4 accumulators, K step = 4.
// A-frag (16x4 fp32, ISA 7.12.2): lane L<16 holds A[m0+L][k..k+1],
//                                 lane L>=16 holds A[m0+L-16][k+2..k+3].
// B-frag (4x16 fp32): lane L<16 holds W[k..k+1][n0+L] (v0,v1),
//                     lane L>=16 holds W[k+2..k+3][n0+L-16].
// C/D (16x16 fp32): VGPR r -> rows r (lanes 0-15) / r+8 (lanes 16-31).
//
// Out-of-range columns: loads are CLAMPED (in-bounds, wrong column) so the
// K-loop is branch-free; WMMA column n only reads B data from the lanes that
// own column n, so polluted accumulator columns are exactly the unstored ones.
// ---------------------------------------------------------------------------
__global__ __launch_bounds__(256) void gemm_logits_kernel(
    const float* __restrict__ A, const float* __restrict__ W,
    float* __restrict__ C, int N) {
  const int lane = threadIdx.x & 31;
  const int wave = threadIdx.x >> 5;
  const int m0 = blockIdx.y * 128 + wave * 16;
  const int n0 = blockIdx.x * 64;
  const int lm = lane & 15;
  const int lh = lane >> 4;  // 0 or 1

  v8f acc0 = {}, acc1 = {}, acc2 = {}, acc3 = {};
  const float* arow = A + (size_t)(m0 + lm) * HIDD;

  const int c0 = n0 + lm, c1 = c0 + 16, c2 = c0 + 32, c3 = c0 + 48;
  // clamped (always in-bounds) load columns; stores stay guarded below
  const int q0 = (c0 < N) ? c0 : (N - 1);
  const int q1 = (c1 < N) ? c1 : (N - 1);
  const int q2 = (c2 < N) ? c2 : (N - 1);
  const int q3 = (c3 < N) ? c3 : (N - 1);

  for (int k = 0; k < HIDD; k += 4) {
    const int ka = k + 2 * lh;                 // even -> 8B-aligned float2
    v2f af = *(const v2f*)(arow + ka);
    __builtin_prefetch(arow + ka + 64, 0, 0);  // global_prefetch_b8, 16 k-steps ahead
    const float* wr0 = W + (size_t)ka * N;
    const float* wr1 = wr0 + N;

    v2f b0 = {wr0[q0], wr1[q0]};
    v2f b1 = {wr0[q1], wr1[q1]};
    v2f b2 = {wr0[q2], wr1[q2]};
    v2f b3 = {wr0[q3], wr1[q3]};

    acc0 = __builtin_amdgcn_wmma_f32_16x16x4_f32(false, af, false, b0, (short)0, acc0, false, false);
    acc1 = __builtin_amdgcn_wmma_f32_16x16x4_f32(false, af, false, b1, (short)0, acc1, false, false);
    acc2 = __builtin_amdgcn_wmma_f32_16x16x4_f32(false, af, false, b2, (short)0, acc2, false, false);
    acc3 = __builtin_amdgcn_wmma_f32_16x16x4_f32(false, af, false, b3, (short)0, acc3, false, false);
  }

#pragma unroll
  for (int r = 0; r < 8; ++r) {
    const size_t row = (size_t)(m0 + r + 8 * lh) * N;
    if (c0 < N) C[row + c0] = acc0[r];
    if (c1 < N) C[row + c1] = acc1[r];
    if (c2 < N) C[row + c2] = acc2[r];
    if (c3 < N) C[row + c3] = acc3[r];
  }
}

// ---------------------------------------------------------------------------
// Phase B: per (b,g) soft-VM. Block = one batch b, wave w = candidate g.
// lane == register index (NR==32) == program step index (T==32).
// ---------------------------------------------------------------------------
__global__ __launch_bounds__(256) void neural_exec_kernel(
    const float* __restrict__ lR, const float* __restrict__ lop,
    const float* __restrict__ ldst, const float* __restrict__ ls1,
    const float* __restrict__ ls2, const float* __restrict__ llen,
    float* __restrict__ Rf) {
  __shared__ float s[816];  // 0:lR[256] 256:lop[64] 320:ldst[160] 480:ls1[160] 640:ls2[160] 800:llen[5]
  const int b = blockIdx.x;
  const int tid = threadIdx.x;

  for (int i = tid; i < NREG * NBIT; i += 256) s[i]       = lR[(size_t)b * 256 + i];
  for (int i = tid; i < TLEN * OPB;  i += 256) s[256 + i] = lop[(size_t)b * 64 + i];
  for (int i = tid; i < TLEN * AB;   i += 256) s[320 + i] = ldst[(size_t)b * 160 + i];
  for (int i = tid; i < TLEN * AB;   i += 256) s[480 + i] = ls1[(size_t)b * 160 + i];
  for (int i = tid; i < TLEN * AB;   i += 256) s[640 + i] = ls2[(size_t)b * 160 + i];
  if (tid < LB) s[800 + tid] = llen[(size_t)b * LB + tid];
  __syncthreads();

  const int lane = tid & 31;
  const int g = tid >> 5;
  const float delta = -1.0f + (2.0f / 7.0f) * (float)g;

  // decode: register file (lane = register), program fields (lane = step)
  float R = 0.0f, M = 0.0f;
#pragma unroll
  for (int i = 0; i < NBIT; ++i)
    R += (float)(1 << i) * sigmoidf(s[lane * NBIT + i] + delta);

  float opd = sigmoidf(s[256 + lane * 2] + delta)
            + 2.0f * sigmoidf(s[256 + lane * 2 + 1] + delta);
  float dstd = 0.0f, s1d = 0.0f, s2d = 0.0f;
#pragma unroll
  for (int i = 0; i < AB; ++i) {
    const float p = (float)(1 << i);
    dstd += p * sigmoidf(s[320 + lane * AB + i] + delta);
    s1d  += p * sigmoidf(s[480 + lane * AB + i] + delta);
    s2d  += p * sigmoidf(s[640 + lane * AB + i] + delta);
  }
  float plen = 0.0f;
#pragma unroll
  for (int i = 0; i < LB; ++i)
    plen += (float)(1 << i) * sigmoidf(s[800 + i] + delta);
  const float act_l = sigmoidf(plen - (float)lane - 0.5f);

  const float fl = (float)lane;
  for (int t = 0; t < TLEN; ++t) {
    const float od  = __shfl(opd,  t, 32);
    const float dd  = __shfl(dstd, t, 32);
    const float a1  = __shfl(s1d,  t, 32);
    const float a2  = __shfl(s2d,  t, 32);
    const float act = __shfl(act_l, t, 32);

    // op softmax over 4 slots (redundant per lane, cheap)
    const float e0 = __expf(-(od - 0.0f) * (od - 0.0f));
    const float e1 = __expf(-(od - 1.0f) * (od - 1.0f));
    const float e2 = __expf(-(od - 2.0f) * (od - 2.0f));
    const float e3 = __expf(-(od - 3.0f) * (od - 3.0f));
    const float ei = 1.0f / (e0 + e1 + e2 + e3);
    const float o0 = e0 * ei, o1 = e1 * ei, o2 = e2 * ei, o3 = e3 * ei;

    // soft one-hot addresses over 32 registers (lane = slot)
    float wd = __expf(-(dd - fl) * (dd - fl)); wd /= wred_sum(wd);
    float w1 = __expf(-(a1 - fl) * (a1 - fl)); w1 /= wred_sum(w1);
    float w2 = __expf(-(a2 - fl) * (a2 - fl)); w2 /= wred_sum(w2);

    const float v1 = wred_sum(R * w1);
    const float v2 = wred_sum(R * w2);
    const float dv = wred_sum(R * wd);
    const float lv = wred_sum(M * w1);

    const float res = o0 * (v1 + v2) + o1 * (v1 - v2) + o2 * lv + o3 * dv;
    const float gr = act * (o0 + o1 + o2) * wd;
    R = R * (1.0f - gr) + gr * res;
    const float gm = act * o3 * wd;
    M = M * (1.0f - gm) + gm * v1;
  }

  Rf[((size_t)b * GNUM + g) * NREG + lane] = R;
}

// ---------------------------------------------------------------------------
// Phase C: out = LayerNorm(Rf[32768,32] @ W_r2h^T[32,512] + b) * g + beta.
// Block = 16 rows; wave w covers cols [w*64, w*64+64) via 4 WMMA tiles (K=32).
// h tile staged in LDS, then per-row LN with wave reductions.
// ---------------------------------------------------------------------------
__global__ __launch_bounds__(256) void r2h_ln_kernel(
    const float* __restrict__ Rf, const float* __restrict__ Wr2h,
    const float* __restrict__ bias, const float* __restrict__ gam,
    const float* __restrict__ beta, float* __restrict__ out) {
  __shared__ float hsh[16 * HIDD];  // 32 KB
  const int lane = threadIdx.x & 31;
  const int wave = threadIdx.x >> 5;
  const int m0 = blockIdx.x * 16;
  const int lm = lane & 15;
  const int lh = lane >> 4;
  const int n0 = wave * 64;

  v8f acc0 = {}, acc1 = {}, acc2 = {}, acc3 = {};
  const float* arow = Rf + (size_t)(m0 + lm) * NREG;

#pragma unroll
  for (int k = 0; k < NREG; k += 4) {
    const int ka = k + 2 * lh;
    v2f af = *(const v2f*)(arow + ka);
    // B[k][n] = Wr2h[n][k]  (torch layout [HID, NR]); consecutive in k -> float2
    v2f b0 = *(const v2f*)(Wr2h + (size_t)(n0 + lm) * NREG + ka);
    v2f b1 = *(const v2f*)(Wr2h + (size_t)(n0 + 16 + lm) * NREG + ka);
    v2f b2 = *(const v2f*)(Wr2h + (size_t)(n0 + 32 + lm) * NREG + ka);
    v2f b3 = *(const v2f*)(Wr2h + (size_t)(n0 + 48 + lm) * NREG + ka);
    acc0 = __builtin_amdgcn_wmma_f32_16x16x4_f32(false, af, false, b0, (short)0, acc0, false, false);
    acc1 = __builtin_amdgcn_wmma_f32_16x16x4_f32(false, af, false, b1, (short)0, acc1, false, false);
    acc2 = __builtin_amdgcn_wmma_f32_16x16x4_f32(false, af, false, b2, (short)0, acc2, false, false);
    acc3 = __builtin_amdgcn_wmma_f32_16x16x4_f32(false, af, false, b3, (short)0, acc3, false, false);
  }

#pragma unroll
  for (int r = 0; r < 8; ++r) {
    const int row = r + 8 * lh;
    const int c = n0 + lm;
    hsh[row * HIDD + c]      = acc0[r] + bias[c];
    hsh[row * HIDD + c + 16] = acc1[r] + bias[c + 16];
    hsh[row * HIDD + c + 32] = acc2[r] + bias[c + 32];
    hsh[row * HIDD + c + 48] = acc3[r] + bias[c + 48];
  }
  __syncthreads();

  // LN: wave w handles rows 2w, 2w+1; each lane owns 16 strided columns
#pragma unroll
  for (int rr = 0; rr < 2; ++rr) {
    const int row = wave * 2 + rr;
    float sum = 0.0f, sq = 0.0f;
#pragma unroll
    for (int i = 0; i < 16; ++i) {
      const float v = hsh[row * HIDD + lane + i * 32];
      sum += v; sq += v * v;
    }
    sum = wred_sum(sum);
    sq  = wred_sum(sq);
    const float mu  = sum * (1.0f / 512.0f);
    const float var = sq * (1.0f / 512.0f) - mu * mu;
    const float inv = rsqrtf(var + 1e-5f);
    float* orow = out + (size_t)(m0 + row) * HIDD;
#pragma unroll
    for (int i = 0; i < 16; ++i) {
      const int c = lane + i * 32;
      const float v = hsh[row * HIDD + c];
      orow[c] = (v - mu) * inv * gam[c] + beta[c];
    }
  }
}

// ---------------------------------------------------------------------------
extern "C" void kernel_launch(void* const* d_in, const int* in_sizes, int n_in,
                              void* d_out, int out_size, void* d_ws, size_t ws_size,
                              hipStream_t stream) {
  (void)in_sizes; (void)n_in; (void)out_size; (void)ws_size;

  const float* z      = (const float*)d_in[0];
  const float* W_R    = (const float*)d_in[1];
  const float* W_op   = (const float*)d_in[2];
  const float* W_dst  = (const float*)d_in[3];
  const float* W_src1 = (const float*)d_in[4];
  const float* W_src2 = (const float*)d_in[5];
  const float* W_len  = (const float*)d_in[6];
  const float* W_r2h  = (const float*)d_in[7];
  const float* b_r2h  = (const float*)d_in[8];
  const float* ln_g   = (const float*)d_in[9];
  const float* ln_b   = (const float*)d_in[10];

  // workspace carve-out (floats): ~17.4 MB total
  float* ws = (float*)d_ws;
  size_t off = 0;
  float* lR   = ws + off; off += (size_t)B_SZ * 256;
  float* lop  = ws + off; off += (size_t)B_SZ * 64;
  float* ldst = ws + off; off += (size_t)B_SZ * 160;
  float* ls1  = ws + off; off += (size_t)B_SZ * 160;
  float* ls2  = ws + off; off += (size_t)B_SZ * 160;
  float* llen = ws + off; off += (size_t)B_SZ * 8;   // stride-5 data, padded region
  float* Rf   = ws + off; off += (size_t)B_SZ * GNUM * NREG;

  const dim3 blk(256);

  // Phase A: six logits GEMMs (one wave = 16x64 C tile)
  gemm_logits_kernel<<<dim3((256 + 63) / 64, B_SZ / 128), blk, 0, stream>>>(z, W_R,    lR,   256);
  gemm_logits_kernel<<<dim3((64  + 63) / 64, B_SZ / 128), blk, 0, stream>>>(z, W_op,   lop,  64);
  gemm_logits_kernel<<<dim3((160 + 63) / 64, B_SZ / 128), blk, 0, stream>>>(z, W_dst,  ldst, 160);
  gemm_logits_kernel<<<dim3((160 + 63) / 64, B_SZ / 128), blk, 0, stream>>>(z, W_src1, ls1,  160);
  gemm_logits_kernel<<<dim3((160 + 63) / 64, B_SZ / 128), blk, 0, stream>>>(z, W_src2, ls2,  160);
  gemm_logits_kernel<<<dim3((5   + 63) / 64, B_SZ / 128), blk, 0, stream>>>(z, W_len,  llen, 5);

  // Phase B: soft-VM scan, block = batch, wave = candidate g
  neural_exec_kernel<<<dim3(B_SZ), blk, 0, stream>>>(lR, lop, ldst, ls1, ls2, llen, Rf);

  // Phase C: register2hidden + LayerNorm, fused
  r2h_ln_kernel<<<dim3((B_SZ * GNUM) / 16), blk, 0, stream>>>(
      Rf, W_r2h, b_r2h, ln_g, ln_b, (float*)d_out);
}